// SelfAttention_81071802679965
// MI455X (gfx1250) — compile-verified
//
#include <hip/hip_runtime.h>
#include <hip/hip_bf16.h>

// Problem constants from the reference setup.
#define B_ 2
#define L_ 512
#define H_ 256
#define NEG_INF_ -1e30f
#define PAD 260   // LDS row stride in floats: multiple of 4 (b128 align), 260 % 64 != 0 -> no bank conflicts

typedef __attribute__((ext_vector_type(2))) float v2f;
typedef __attribute__((ext_vector_type(8))) float v8f;

__device__ __forceinline__ float fast_tanh(float x) {
#if __has_builtin(__builtin_amdgcn_tanhf)
    return __builtin_amdgcn_tanhf(x);            // gfx1250 v_tanh_f32 (TRANS pipe)
#else
    float e = __builtin_amdgcn_exp2f(x * 2.8853900817779268f);  // exp(2x)
    return (e - 1.0f) / (e + 1.0f);
#endif
}

// ---------------------------------------------------------------------------
// Pass 1: PS = c*Wself^T + bself ; PO = c*Wother^T + bother
// One wave per 16x16 output tile, K-loop of V_WMMA_F32_16X16X4_F32.
// A[m][k]=c[r0+m][h0+k]; B[k][n]=W[o0+n][h0+k] (both contiguous float2 per lane).
// ---------------------------------------------------------------------------
__global__ void proj_wmma(const float* __restrict__ c,
                          const float* __restrict__ Wself, const float* __restrict__ bself,
                          const float* __restrict__ Wother, const float* __restrict__ bother,
                          float* __restrict__ ps, float* __restrict__ po) {
    const int wave = threadIdx.x >> 5;
    const int lane = threadIdx.x & 31;
    int tile = blockIdx.x * 4 + wave;        // 2048 tiles total
    const int which = tile >> 10;            // 0=self, 1=other
    tile &= 1023;                            // 64 row-tiles x 16 col-tiles
    const int r0 = (tile >> 4) << 4;         // flattened B*L row tile
    const int o0 = (tile & 15) << 4;         // output-feature tile

    const float* W    = which ? Wother : Wself;
    const float* bias = which ? bother : bself;
    float*       out  = which ? po     : ps;

    const int half = lane >> 4;              // K-half select per ISA A/B f32 layout
    const int lm   = lane & 15;              // M (A rows) / N (B cols)

    const float* arow = c + (size_t)(r0 + lm) * H_ + 2 * half;
    const float* brow = W + (size_t)(o0 + lm) * H_ + 2 * half;

    v8f acc = {};
    for (int k = 0; k < H_; k += 4) {
        v2f a = *(const v2f*)(arow + k);
        v2f b = *(const v2f*)(brow + k);
        acc = __builtin_amdgcn_wmma_f32_16x16x4_f32(false, a, false, b,
                                                    (short)0, acc, false, false);
    }
    const float bb = bias[o0 + lm];
    float* orow = out + (size_t)r0 * H_ + o0 + lm;
#pragma unroll
    for (int r = 0; r < 8; ++r)              // D layout: vgpr r holds M = r + 8*half, N = lm
        orow[(size_t)(r + 8 * half) * H_] = acc[r] + bb;
}

// ---------------------------------------------------------------------------
// Pass 2: s[b,i,j] = sum_h tanh(PS[b,i,h] + PO[b,j,h]) * v[h]
// 16x16 (i,j) tile per 256-thread block; PS/PO/v staged in LDS (padded rows).
// This is the TRANS-bound hot loop: 256 tanh+fma per thread.
// ---------------------------------------------------------------------------
__global__ void score_tanh(const float* __restrict__ ps, const float* __restrict__ po,
                           const float* __restrict__ v, float* __restrict__ s) {
    __shared__ float sm_ps[16 * PAD];
    __shared__ float sm_po[16 * PAD];
    __shared__ float sm_v[H_];

    int t = blockIdx.x;                      // 2048 = B * 32 * 32
    const int b = t >> 10; t &= 1023;
    const int i0 = (t >> 5) << 4;
    const int j0 = (t & 31) << 4;

    const int tid = threadIdx.x;
    sm_v[tid] = v[tid];                      // H_ == blockDim.x == 256
    for (int idx = tid; idx < 16 * H_; idx += 256) {
        const int row = idx >> 8, col = idx & (H_ - 1);
        sm_ps[row * PAD + col] = ps[((size_t)b * L_ + i0 + row) * H_ + col];
        sm_po[row * PAD + col] = po[((size_t)b * L_ + j0 + row) * H_ + col];
    }
    __syncthreads();

    const int i = tid >> 4, j = tid & 15;
    const float* pr = sm_ps + i * PAD;
    const float* qr = sm_po + j * PAD;

    float acc = 0.0f;
#pragma unroll 4
    for (int h = 0; h < H_; h += 4) {        // ds_load_b128 x3 per 4 elements
        float4 a  = *(const float4*)(pr + h);
        float4 bb = *(const float4*)(qr + h);
        float4 vv = *(const float4*)(sm_v + h);
        acc += fast_tanh(a.x + bb.x) * vv.x;
        acc += fast_tanh(a.y + bb.y) * vv.y;
        acc += fast_tanh(a.z + bb.z) * vv.z;
        acc += fast_tanh(a.w + bb.w) * vv.w;
    }
    s[((size_t)b * L_ + i0 + i) * L_ + j0 + j] = acc;
}

// ---------------------------------------------------------------------------
// Pass 3: masked softmax over the key axis (L=512), one block per row.
// ---------------------------------------------------------------------------
__global__ void softmax_rows(float* __restrict__ s, const unsigned char* __restrict__ mask) {
    __shared__ float red[256];
    const int row = blockIdx.x;              // 0 .. B*L-1
    const int b   = row / L_;
    const int tid = threadIdx.x;
    float* srow = s + (size_t)row * L_;
    const unsigned char* mrow = mask + (size_t)b * L_;

    float x0 = mrow[tid]       ? srow[tid]       : NEG_INF_;
    float x1 = mrow[tid + 256] ? srow[tid + 256] : NEG_INF_;

    red[tid] = fmaxf(x0, x1);
    __syncthreads();
    for (int off = 128; off > 0; off >>= 1) {
        if (tid < off) red[tid] = fmaxf(red[tid], red[tid + off]);
        __syncthreads();
    }
    const float m = red[0];
    __syncthreads();

    const float LOG2E = 1.4426950408889634f;
    float p0 = __builtin_amdgcn_exp2f((x0 - m) * LOG2E);
    float p1 = __builtin_amdgcn_exp2f((x1 - m) * LOG2E);
    red[tid] = p0 + p1;
    __syncthreads();
    for (int off = 128; off > 0; off >>= 1) {
        if (tid < off) red[tid] += red[tid + off];
        __syncthreads();
    }
    const float inv = 1.0f / red[0];
    srow[tid]       = p0 * inv;
    srow[tid + 256] = p1 * inv;
}

// ---------------------------------------------------------------------------
// Pass 4: out[b] = A[b] (LxL) * c[b] (LxH) with V_WMMA_F32_16X16X4_F32.
// A[m][k]=P[i0+m][j0+k] (float2 per lane); B[k][n]=c[j0+k][h0+n] (2 scalars).
// ---------------------------------------------------------------------------
__global__ void av_wmma(const float* __restrict__ p, const float* __restrict__ c,
                        float* __restrict__ out) {
    const int wave = threadIdx.x >> 5;
    const int lane = threadIdx.x & 31;
    int tile = blockIdx.x * 4 + wave;        // 1024 tiles = B * 32 * 16
    const int b = tile >> 9; tile &= 511;
    const int i0 = (tile >> 4) << 4;
    const int h0 = (tile & 15) << 4;

    const int half = lane >> 4;
    const int lm   = lane & 15;

    const float* prow = p + ((size_t)b * L_ + i0 + lm) * L_ + 2 * half;
    const float* cb   = c + (size_t)b * L_ * H_ + h0 + lm;

    v8f acc = {};
    for (int k = 0; k < L_; k += 4) {
        v2f a = *(const v2f*)(prow + k);
        v2f bb;
        bb.x = cb[(size_t)(k + 2 * half)     * H_];
        bb.y = cb[(size_t)(k + 2 * half + 1) * H_];
        acc = __builtin_amdgcn_wmma_f32_16x16x4_f32(false, a, false, bb,
                                                    (short)0, acc, false, false);
    }
    float* orow = out + ((size_t)b * L_ + i0) * H_ + h0 + lm;
#pragma unroll
    for (int r = 0; r < 8; ++r)
        orow[(size_t)(r + 8 * half) * H_] = acc[r];
}

// ---------------------------------------------------------------------------
extern "C" void kernel_launch(void* const* d_in, const int* in_sizes, int n_in,
                              void* d_out, int out_size, void* d_ws, size_t ws_size,
                              hipStream_t stream) {
    const float*         c    = (const float*)d_in[0];
    const unsigned char* mask = (const unsigned char*)d_in[1];   // numpy bool = 1 byte
    const float*         Wsw  = (const float*)d_in[2];
    const float*         Wsb  = (const float*)d_in[3];
    const float*         Wow  = (const float*)d_in[4];
    const float*         Wob  = (const float*)d_in[5];
    const float*         v    = (const float*)d_in[6];
    float* out = (float*)d_out;

    // Workspace layout: PS (1MB) | PO (1MB) | S (2MB)
    float* ps = (float*)d_ws;
    float* po = ps + (size_t)B_ * L_ * H_;
    float* s  = po + (size_t)B_ * L_ * H_;

    proj_wmma  <<<512,      128, 0, stream>>>(c, Wsw, Wsb, Wow, Wob, ps, po);
    score_tanh <<<2048,     256, 0, stream>>>(ps, po, v, s);
    softmax_rows<<<B_ * L_, 256, 0, stream>>>(s, mask);
    av_wmma    <<<256,      128, 0, stream>>>(s, c, out);
}